// MultiHeadAttn_63660005262000
// MI455X (gfx1250) — compile-verified
//
#include <hip/hip_runtime.h>

// MHA forward for gfx1250 (MI455X): fp32 WMMA (V_WMMA_F32_16X16X4_F32) throughout.
// Layouts per CDNA5 ISA (wave32):
//   A 16x4 f32 : lane&15 = M, (lane>>4)*2+{0,1} = K  -> v2f
//   B 4x16 f32 : lane&15 = N, (lane>>4)*2+{0,1} = K  -> v2f
//   C/D 16x16  : vgpr i -> M = i + (lane>>4)*8, N = lane&15 -> v8f

typedef float v2f __attribute__((ext_vector_type(2)));
typedef float v8f __attribute__((ext_vector_type(8)));

#define SEQ    2048
#define BATCH  2
#define NHEAD  8
#define DHEAD  64
#define DMODEL 512
#define MROWS  (SEQ * BATCH)   // 4096 flattened (s,b) rows

static __device__ __forceinline__ v8f wmma_f32(v2f a, v2f b, v8f c) {
    return __builtin_amdgcn_wmma_f32_16x16x4_f32(false, a, false, b, (short)0, c, false, false);
}

// ---------------------------------------------------------------------------
// Kernel 1: QKV projection.  dst[b][h][s][d] = scale * sum_k X[s*B+b,k] * W[h*64+d,k]
// Block = 256 threads = 8 waves; wave tile 16x16; block tile 32(M) x 64(N).
// ---------------------------------------------------------------------------
__global__ __launch_bounds__(256) void proj_qkv(const float* __restrict__ X,   // [4096,512]
                                                const float* __restrict__ W,   // [512,512] (pre-offset)
                                                float* __restrict__ dst,       // [B,H,S,64]
                                                float scale)
{
    const int lane = threadIdx.x & 31;
    const int wave = threadIdx.x >> 5;
    const int m0 = blockIdx.x * 32 + (wave & 1) * 16;
    const int n0 = blockIdx.y * 64 + (wave >> 1) * 16;
    const int l15 = lane & 15;
    const int hi  = lane >> 4;

    const float* arow = X + (size_t)(m0 + l15) * DMODEL;
    const float* brow = W + (size_t)(n0 + l15) * DMODEL;   // B[k,n] = W[n,k]

    v8f acc = {};
#pragma unroll 4
    for (int k0 = 0; k0 < DMODEL; k0 += 4) {
        const int kb = k0 + hi * 2;
        v2f a, b;
        a.x = arow[kb]; a.y = arow[kb + 1];
        b.x = brow[kb]; b.y = brow[kb + 1];
        acc = wmma_f32(a, b, acc);
    }

    const int ncol = n0 + l15;
    const int hh = ncol >> 6, d = ncol & 63;
#pragma unroll
    for (int i = 0; i < 8; ++i) {
        const int r = m0 + i + hi * 8;        // = s*BATCH + b
        const int s = r >> 1, bb = r & 1;
        dst[(((size_t)(bb * NHEAD + hh) * SEQ) + s) * DHEAD + d] = acc[i] * scale;
    }
}

// ---------------------------------------------------------------------------
// Kernel 2: causal flash attention per (b,h).  One wave = one 16-row Q tile.
// Q is pre-scaled by 1/sqrt(DHEAD); masked scores = -1e6 (matches reference).
// ---------------------------------------------------------------------------
__global__ __launch_bounds__(256) void flash_attn(const float* __restrict__ Q,  // [B,H,S,64]
                                                  const float* __restrict__ K,
                                                  const float* __restrict__ V,
                                                  float* __restrict__ O)        // [B,H,S,64]
{
    __shared__ float pbuf[8 * 16 * 16];       // per-wave 16x16 P staging
    const int lane = threadIdx.x & 31;
    const int wave = threadIdx.x >> 5;
    const int task = blockIdx.x * 8 + wave;   // 2048 wave tasks
    const int bh = task >> 7;                 // 0..15
    const int qt = task & 127;                // q tile index
    const int q0 = qt * 16;
    const size_t base = (size_t)bh * SEQ * DHEAD;
    const int l15 = lane & 15;
    const int hi  = lane >> 4;
    float* pl = pbuf + wave * 256;

    // Preload Q A-fragments: 16 chunks over d (K-dim of S = Q*K^T)
    v2f qa[16];
    {
        const float* qrow = Q + base + (size_t)(q0 + l15) * DHEAD;
#pragma unroll
        for (int dc = 0; dc < 16; ++dc) {
            const int kb = dc * 4 + hi * 2;
            qa[dc].x = qrow[kb]; qa[dc].y = qrow[kb + 1];
        }
    }

    v8f o0 = {}, o1 = {}, o2 = {}, o3 = {};
    float mrow[8], lrow[8];
#pragma unroll
    for (int i = 0; i < 8; ++i) { mrow[i] = -3.0e38f; lrow[i] = 0.0f; }

    for (int kc = 0; kc <= qt; ++kc) {        // causal: only k-tiles <= q-tile
        const int k0 = kc * 16;

        // S-tile = Q(16x64) * K^T(64x16)
        v8f s = {};
        const float* krow = K + base + (size_t)(k0 + l15) * DHEAD; // B[d, kcol]=K[kcol,d]
#pragma unroll
        for (int dc = 0; dc < 16; ++dc) {
            const int kb = dc * 4 + hi * 2;
            v2f b; b.x = krow[kb]; b.y = krow[kb + 1];
            s = wmma_f32(qa[dc], b, s);
        }

        // Causal mask + online softmax (row reductions within each 16-lane half)
        float alpha[8];
#pragma unroll
        for (int i = 0; i < 8; ++i) {
            const int qg = q0 + i + hi * 8;
            const int kg = k0 + l15;
            float sv = (kg > qg) ? -1.0e6f : s[i];
            float rm = sv;
#pragma unroll
            for (int msk = 1; msk <= 8; msk <<= 1)
                rm = fmaxf(rm, __shfl_xor(rm, msk, 32));
            const float mnew = fmaxf(mrow[i], rm);
            const float p = __expf(sv - mnew);
            float rs = p;
#pragma unroll
            for (int msk = 1; msk <= 8; msk <<= 1)
                rs += __shfl_xor(rs, msk, 32);
            alpha[i] = __expf(mrow[i] - mnew);
            lrow[i] = lrow[i] * alpha[i] + rs;
            mrow[i] = mnew;
            pl[(i + hi * 8) * 16 + l15] = p;   // P in row-major LDS tile
        }
        __builtin_amdgcn_wave_barrier();

        // Rescale running O by alpha
#pragma unroll
        for (int i = 0; i < 8; ++i) {
            o0[i] *= alpha[i]; o1[i] *= alpha[i];
            o2[i] *= alpha[i]; o3[i] *= alpha[i];
        }

        // P C-layout -> A-layout via LDS
        v2f pa[4];
#pragma unroll
        for (int pc = 0; pc < 4; ++pc) {
            const int kk = pc * 4 + hi * 2;
            pa[pc].x = pl[l15 * 16 + kk];
            pa[pc].y = pl[l15 * 16 + kk + 1];
        }
        __builtin_amdgcn_wave_barrier();

        // O(16x64) += P(16x16) * V(16x64)
        const float* vbase = V + base + (size_t)k0 * DHEAD;
#pragma unroll
        for (int pc = 0; pc < 4; ++pc) {
            const int r0 = pc * 4 + hi * 2;
            const float* vr0 = vbase + (size_t)r0 * DHEAD + l15;
            const float* vr1 = vbase + (size_t)(r0 + 1) * DHEAD + l15;
            v2f vb;
            vb.x = vr0[0];  vb.y = vr1[0];  o0 = wmma_f32(pa[pc], vb, o0);
            vb.x = vr0[16]; vb.y = vr1[16]; o1 = wmma_f32(pa[pc], vb, o1);
            vb.x = vr0[32]; vb.y = vr1[32]; o2 = wmma_f32(pa[pc], vb, o2);
            vb.x = vr0[48]; vb.y = vr1[48]; o3 = wmma_f32(pa[pc], vb, o3);
        }
    }

    // Normalize and store
#pragma unroll
    for (int i = 0; i < 8; ++i) {
        const float inv = 1.0f / lrow[i];
        float* orow = O + base + (size_t)(q0 + i + hi * 8) * DHEAD;
        orow[ 0 + l15] = o0[i] * inv;
        orow[16 + l15] = o1[i] * inv;
        orow[32 + l15] = o2[i] * inv;
        orow[48 + l15] = o3[i] * inv;
    }
}

// ---------------------------------------------------------------------------
// Kernel 3: output projection.  out[s,b,n] = sum_f attn[b][f>>6][s][f&63] * Wo[n,f]
// ---------------------------------------------------------------------------
__global__ __launch_bounds__(256) void out_proj(const float* __restrict__ Av,  // [B,H,S,64]
                                                const float* __restrict__ Wo,  // [512,512]
                                                float* __restrict__ out)       // [S,B,512]
{
    const int lane = threadIdx.x & 31;
    const int wave = threadIdx.x >> 5;
    const int m0 = blockIdx.x * 32 + (wave & 1) * 16;
    const int n0 = blockIdx.y * 64 + (wave >> 1) * 16;
    const int l15 = lane & 15;
    const int hi  = lane >> 4;

    const int arow = m0 + l15;                 // = s*BATCH + b
    const int s = arow >> 1, bb = arow & 1;
    const int ncol = n0 + l15;
    const float* wrow = Wo + (size_t)ncol * DMODEL;

    v8f acc = {};
#pragma unroll 4
    for (int k0 = 0; k0 < DMODEL; k0 += 4) {
        const int f0 = k0 + hi * 2;
        const int f1 = f0 + 1;
        v2f a, b;
        a.x = Av[((size_t)(bb * NHEAD + (f0 >> 6)) * SEQ + s) * DHEAD + (f0 & 63)];
        a.y = Av[((size_t)(bb * NHEAD + (f1 >> 6)) * SEQ + s) * DHEAD + (f1 & 63)];
        b.x = wrow[f0]; b.y = wrow[f1];
        acc = wmma_f32(a, b, acc);
    }

#pragma unroll
    for (int i = 0; i < 8; ++i) {
        const int r = m0 + i + hi * 8;         // out flat row = s*BATCH+b
        out[(size_t)r * DMODEL + (n0 + l15)] = acc[i];
    }
}

// ---------------------------------------------------------------------------
extern "C" void kernel_launch(void* const* d_in, const int* in_sizes, int n_in,
                              void* d_out, int out_size, void* d_ws, size_t ws_size,
                              hipStream_t stream) {
    const float* h   = (const float*)d_in[0];   // [S,B,512]
    const float* Wq  = (const float*)d_in[1];   // [512,512]
    const float* Wkv = (const float*)d_in[2];   // [1024,512]
    const float* Wo  = (const float*)d_in[3];   // [512,512]
    float* out = (float*)d_out;                 // [S,B,512]

    const size_t hbuf = (size_t)BATCH * NHEAD * SEQ * DHEAD;  // 2 MiB floats = 8 MB
    float* Qb = (float*)d_ws;
    float* Kb = Qb + hbuf;
    float* Vb = Kb + hbuf;
    float* Ab = Vb + hbuf;

    dim3 gemmGrid(MROWS / 32, DMODEL / 64);     // (128, 8)
    const float qscale = 0.125f;                // 1/sqrt(DHEAD), folded into Q

    proj_qkv<<<gemmGrid, 256, 0, stream>>>(h, Wq,                     Qb, qscale);
    proj_qkv<<<gemmGrid, 256, 0, stream>>>(h, Wkv,                    Kb, 1.0f);
    proj_qkv<<<gemmGrid, 256, 0, stream>>>(h, Wkv + (size_t)512 * 512, Vb, 1.0f);

    flash_attn<<<SEQ * BATCH * NHEAD / (16 * 8), 256, 0, stream>>>(Qb, Kb, Vb, Ab);

    out_proj<<<gemmGrid, 256, 0, stream>>>(Ab, Wo, out);
}